// RMSNorm_24455543783487
// MI455X (gfx1250) — compile-verified
//
#include <hip/hip_runtime.h>
#include <hip/hip_bf16.h>
#include <stdint.h>

// RMSNorm over last dim: x (4,4096,2048) fp32, gain (2048) fp32.
// Bandwidth-bound: ~268 MB traffic -> ~11.5 us floor at 23.3 TB/s.
// Strategy: async global->LDS staging (CDNA5 data mover), double-buffered per
// block, wave32 shuffle reduction, nontemporal 128-bit stores.

#define D_MODEL        2048
#define RMS_EPS        1e-5f
#define BLOCK          256
#define VEC_PER_ROW    (D_MODEL / 4)          // 512 float4 per row
#define ROWS_PER_BLOCK 8

typedef float v4f __attribute__((ext_vector_type(4)));

// Builtin-exact types: gcc-style int4 vector (matches clang's V4i builtin type),
// in the global (AS1) and LDS (AS3) address spaces.
typedef int v4i_gcc __attribute__((vector_size(16)));
typedef __attribute__((address_space(1))) v4i_gcc g_v4i;
typedef __attribute__((address_space(3))) v4i_gcc l_v4i;

// ---- CDNA5 async global->LDS 128-bit load ------------------------------
__device__ __forceinline__ void async_load_b128(const float* gsrc, const v4f* ldst) {
#if __has_builtin(__builtin_amdgcn_global_load_async_to_lds_b128)
    __builtin_amdgcn_global_load_async_to_lds_b128(
        (g_v4i*)(uintptr_t)gsrc,
        (l_v4i*)(uint32_t)(uintptr_t)ldst,
        /*imm offset*/0, /*cpol*/0);
#else
    uint32_t lds_off = (uint32_t)(uintptr_t)ldst;     // low 32 bits of generic LDS ptr = DS address
    uint64_t gaddr   = (uint64_t)(uintptr_t)gsrc;
    asm volatile("global_load_async_to_lds_b128 %0, %1, off"
                 :: "v"(lds_off), "v"(gaddr)
                 : "memory");
#endif
}

__device__ __forceinline__ void wait_async_le2() {
#if __has_builtin(__builtin_amdgcn_s_wait_asynccnt)
    __builtin_amdgcn_s_wait_asynccnt(2);
#else
    asm volatile("s_wait_asynccnt 2" ::: "memory");
#endif
}

__device__ __forceinline__ void wait_async_0() {
#if __has_builtin(__builtin_amdgcn_s_wait_asynccnt)
    __builtin_amdgcn_s_wait_asynccnt(0);
#else
    asm volatile("s_wait_asynccnt 0" ::: "memory");
#endif
}

// ------------------------------------------------------------------------
__global__ __launch_bounds__(BLOCK)
void RMSNorm_24455543783487_kernel(const float* __restrict__ x,
                                   const float* __restrict__ gain,
                                   float* __restrict__ out,
                                   int nrows) {
    __shared__ v4f  buf[2][VEC_PER_ROW];      // 2 x 8 KB double buffer
    __shared__ float wsum[BLOCK / 32];        // one partial per wave32

    const int t    = threadIdx.x;
    const int lane = t & 31;
    const int wave = t >> 5;

    const int base_row = blockIdx.x * ROWS_PER_BLOCK;
    int count = nrows - base_row;
    if (count <= 0) return;
    if (count > ROWS_PER_BLOCK) count = ROWS_PER_BLOCK;

    // Gain is reused for every row this block processes: load once.
    const v4f* g4 = (const v4f*)gain;
    const v4f ga = g4[t];
    const v4f gb = g4[t + BLOCK];

    // Prologue: stage row 0 into buf[0] (each wave covers its own lanes' slots).
    {
        const float* src = x + (size_t)base_row * D_MODEL;
        async_load_b128(src + t * 4,            &buf[0][t]);
        async_load_b128(src + (t + BLOCK) * 4,  &buf[0][t + BLOCK]);
    }

    for (int i = 0; i < count; ++i) {
        const int cur = i & 1;

        if (i + 1 < count) {
            // Prefetch next row into the other buffer, then wait for only the
            // current row's two loads (leave the 2 prefetch ops in flight).
            const float* src = x + (size_t)(base_row + i + 1) * D_MODEL;
            async_load_b128(src + t * 4,           &buf[1 - cur][t]);
            async_load_b128(src + (t + BLOCK) * 4, &buf[1 - cur][t + BLOCK]);
            wait_async_le2();
        } else {
            wait_async_0();
        }
        __syncthreads();

        // Each thread reads back its own 8 elements (ds_load_b128 x2).
        const v4f a = buf[cur][t];
        const v4f b = buf[cur][t + BLOCK];

        float s = a.x * a.x + a.y * a.y + a.z * a.z + a.w * a.w
                + b.x * b.x + b.y * b.y + b.z * b.z + b.w * b.w;

        // wave32 butterfly reduction
        #pragma unroll
        for (int off = 16; off > 0; off >>= 1)
            s += __shfl_xor(s, off, 32);

        if (lane == 0) wsum[wave] = s;
        __syncthreads();

        float total = 0.0f;
        #pragma unroll
        for (int w = 0; w < BLOCK / 32; ++w) total += wsum[w];

        const float inv = 1.0f / sqrtf(total * (1.0f / (float)D_MODEL) + RMS_EPS);

        // Scale + gain, stream out with non-temporal 128-bit stores.
        float* dst = out + (size_t)(base_row + i) * D_MODEL;
        const v4f oa = a * inv * ga;
        const v4f ob = b * inv * gb;
        __builtin_nontemporal_store(oa, (v4f*)(dst + t * 4));
        __builtin_nontemporal_store(ob, (v4f*)(dst + (t + BLOCK) * 4));

        __syncthreads();   // protect wsum (and buffer rotation) for next row
    }
}

// ------------------------------------------------------------------------
extern "C" void kernel_launch(void* const* d_in, const int* in_sizes, int n_in,
                              void* d_out, int out_size, void* d_ws, size_t ws_size,
                              hipStream_t stream) {
    const float* x    = (const float*)d_in[0];
    const float* gain = (const float*)d_in[1];
    float*       out  = (float*)d_out;

    const int nrows  = in_sizes[0] / D_MODEL;   // 4*4096 = 16384
    const int blocks = (nrows + ROWS_PER_BLOCK - 1) / ROWS_PER_BLOCK;

    RMSNorm_24455543783487_kernel<<<blocks, BLOCK, 0, stream>>>(x, gain, out, nrows);
}